// GlobalProdRenormNormalConstCovStdPrior_53291954209210
// MI455X (gfx1250) — compile-verified
//
#include <hip/hip_runtime.h>

// GlobalProdRenormNormalConstCovStdPrior for MI455X (gfx1250, wave32).
//
// Restructured math:
//   prec_b = I + N_b * prec_i  (SPD)
//   reverse-Cholesky: prec_b = V V^T, V upper (factored from bottom-right)
//   U = V^{-1} (upper)  =>  cov_b = U^T U, and U is EXACTLY the reference's
//   upper Cholesky factor of cov_b.  So: chol_out = U / diag(U) (rows),
//   logvar = 2 log diag(U), mu = U^T (U gamma).  cov is never materialized.
//   prec_i itself comes from the same routine run once on cov_i.
// All O(D^3) work = 16x16x16 tile GEMMs on V_WMMA_F32_16X16X4_F32.

#define Dm     512
#define NB     32          // 512 / 16 blocks
#define NBATCH 256
#define Tt     300

typedef float v2f __attribute__((ext_vector_type(2)));
typedef float v8f __attribute__((ext_vector_type(8)));

static __device__ __forceinline__ v8f vzero() {
  v8f z = {0.f, 0.f, 0.f, 0.f, 0.f, 0.f, 0.f, 0.f};
  return z;
}

// intra-wave LDS fence (lockstep wave32 + in-order LDS; keep compiler honest)
static __device__ __forceinline__ void wave_fence() {
  __builtin_amdgcn_wave_barrier();
  asm volatile("s_wait_dscnt 0" ::: "memory");
}

// One 16x16x16 tile-GEMM: acc += A(16x16) * B(16x16) via 4x WMMA f32 16x16x4.
// la(m,k) returns A[m][k]; lb(k,n) returns B[k][n].
// A frag: lanes0-15 hold K=k0,k0+1 ; lanes16-31 hold K=k0+2,k0+3 (ISA layout).
template <typename FA, typename FB>
static __device__ __forceinline__ v8f wmma16(const FA& la, const FB& lb, v8f acc) {
  const int lane = threadIdx.x & 31;
  const int hi2  = (lane >> 4) << 1;  // 0 or 2
  const int ln   = lane & 15;
#pragma unroll
  for (int k0 = 0; k0 < 16; k0 += 4) {
    v2f a, b;
    a.x = la(ln, k0 + hi2);
    a.y = la(ln, k0 + hi2 + 1);
    b.x = lb(k0 + hi2, ln);
    b.y = lb(k0 + hi2 + 1, ln);
    acc = __builtin_amdgcn_wmma_f32_16x16x4_f32(false, a, false, b,
                                                (short)0, acc, false, false);
  }
  return acc;
}

// U element accessor after TRTRI: U^T lives in the strictly-lower half of Mb,
// inverted diagonal blocks live in LDS (dall[32][16][16]).
static __device__ __forceinline__ float loadU(const float* Mb, const float* dall,
                                              int r, int c) {
  if (c < r) return 0.f;
  const int br = r >> 4, bc = c >> 4;
  if (br == bc) return dall[br * 256 + (r & 15) * 16 + (c & 15)];
  return Mb[c * Dm + r];
}

// ---------------------------------------------------------------------------
// Blocked reverse Cholesky: A = V V^T (V upper), in the upper triangle of Mb.
// After each diagonal block is factored, its INVERSE is written back into the
// diagonal block of Mb (consumed later by trtri512).  256 threads = 8 waves.
// ---------------------------------------------------------------------------
static __device__ void rchol512(float* Mb, float* pan, float* dg, float* dinv,
                                int tid) {
  const int wid = tid >> 5;
  const int hi8 = ((tid & 31) >> 4) << 3;
  const int ln  = tid & 15;

  for (int kb = NB - 1; kb >= 0; --kb) {
    // 1) load diagonal block
    {
      int r = tid >> 4, c = tid & 15;
      dg[r * 17 + c] = Mb[(kb * 16 + r) * Dm + kb * 16 + c];
    }
    __syncthreads();

    // 2) reverse Cholesky of the 16x16 block (columns right-to-left)
    for (int c = 15; c >= 0; --c) {
      if (tid < 16) {
        float sc = 0.f, sr = 0.f;
        for (int k = c + 1; k < 16; ++k) {
          float vck = dg[c * 17 + k];
          sc += vck * vck;
          sr += dg[tid * 17 + k] * vck;
        }
        float vcc = sqrtf(fmaxf(dg[c * 17 + c] - sc, 1e-30f));
        if (tid == c) dg[c * 17 + c] = vcc;
        else if (tid < c) dg[tid * 17 + c] = (dg[tid * 17 + c] - sr) / vcc;
      }
      __syncthreads();
    }

    // 3) invert the upper-triangular block -> dinv (columns left-to-right)
    for (int c = 0; c < 16; ++c) {
      if (tid < 16) {
        float vcc = dg[c * 17 + c];
        float val;
        if (tid == c) val = 1.0f / vcc;
        else if (tid < c) {
          float s = 0.f;
          for (int k = tid; k < c; ++k) s += dinv[tid * 17 + k] * dg[k * 17 + c];
          val = -s / vcc;
        } else val = 0.f;
        dinv[tid * 17 + c] = val;
      }
      __syncthreads();
    }

    // 4) stash inv(V[kb,kb]) into the diagonal block of Mb (V diag not needed)
    {
      int r = tid >> 4, c = tid & 15;
      Mb[(kb * 16 + r) * Dm + kb * 16 + c] = dinv[r * 17 + c];
    }

    // 5) panel: V[i,kb] = A[i,kb] * Dinv^T   (WMMA; to LDS panel + global)
    for (int t = wid; t < kb; t += 8) {
      const float* Ablk = Mb + (t * 16) * Dm + kb * 16;
      v8f acc = vzero();
      acc = wmma16([&](int m, int k) { return Ablk[m * Dm + k]; },
                   [&](int k, int n) { return dinv[n * 17 + k]; }, acc);
      float* pt = pan + t * 256;
      float* gt = Mb + (t * 16) * Dm + kb * 16;
#pragma unroll
      for (int r = 0; r < 8; ++r) {
        pt[(r + hi8) * 16 + ln] = acc[r];
        gt[(r + hi8) * Dm + ln] = acc[r];
      }
    }
    __syncthreads();

    // 6) trailing SYRK update: A[i,j] -= P_i P_j^T for i<=j<kb (WMMA)
    const int ntile = kb * (kb + 1) / 2;
    for (int t = wid; t < ntile; t += 8) {
      int j = 0;
      while ((j + 1) * (j + 2) / 2 <= t) ++j;
      int i = t - j * (j + 1) / 2;
      const float* Pi = pan + i * 256;
      const float* Pj = pan + j * 256;
      v8f acc = vzero();
      acc = wmma16([&](int m, int k) { return Pi[m * 16 + k]; },
                   [&](int k, int n) { return Pj[n * 16 + k]; }, acc);
      float* Cb = Mb + (i * 16) * Dm + j * 16;
#pragma unroll
      for (int r = 0; r < 8; ++r) Cb[(r + hi8) * Dm + ln] -= acc[r];
    }
    __syncthreads();
  }
}

// ---------------------------------------------------------------------------
// Blocked upper-triangular inverse U = V^{-1}.
// Inverted diagonal blocks -> LDS dall.  Off-diagonal U written as U^T into
// the strictly-lower half of Mb (V's upper half stays intact while needed).
// Row i is owned entirely by one wave -> no cross-wave hazards.
//   U[i,j] = -( sum_{k=i}^{j-1} U[i,k] V[k,j] ) * Dinv_j
// ---------------------------------------------------------------------------
static __device__ void trtri512(float* Mb, float* dall, float* stg, int tid) {
  const int wid = tid >> 5;
  const int hi8 = ((tid & 31) >> 4) << 3;
  const int ln  = tid & 15;

  // a) gather the inverted diagonal blocks (stored by rchol512) into LDS
  for (int idx = tid; idx < NB * 256; idx += 256) {
    int blk = idx >> 8, e = idx & 255, r = e >> 4, c = e & 15;
    dall[idx] = Mb[(blk * 16 + r) * Dm + blk * 16 + c];
  }
  __syncthreads();

  float* ws = stg + wid * 256;  // per-wave 16x16 staging tile
  for (int i = wid; i < NB; i += 8) {
    for (int j = i + 1; j < NB; ++j) {
      v8f acc = vzero();
      for (int k = i; k < j; ++k) {
        acc = wmma16(
            [&](int m, int kk) { return loadU(Mb, dall, i * 16 + m, k * 16 + kk); },
            [&](int kk, int n) { return Mb[(k * 16 + kk) * Dm + j * 16 + n]; },
            acc);
      }
      // stage S through LDS to feed it back in as an A-operand
#pragma unroll
      for (int r = 0; r < 8; ++r) ws[(r + hi8) * 16 + ln] = acc[r];
      wave_fence();
      const float* Dj = dall + j * 256;
      v8f res = vzero();
      res = wmma16([&](int m, int kk) { return ws[m * 16 + kk]; },
                   [&](int kk, int n) { return Dj[kk * 16 + n]; }, res);
      // store U[i,j]^T into the strictly-lower block (j,i)
#pragma unroll
      for (int r = 0; r < 8; ++r)
        Mb[(j * 16 + ln) * Dm + i * 16 + (r + hi8)] = -res[r];
      wave_fence();
    }
  }
  __syncthreads();
}

// ---------------------------------------------------------------------------
// kernels
// ---------------------------------------------------------------------------
__global__ void k_var(const float* Dw, float* var) {
  int i = blockIdx.x * blockDim.x + threadIdx.x;
  if (i < Dm) var[i] = expf(Dw[i]);
}

__global__ void k_stats(const float* __restrict__ x, const float* __restrict__ w,
                        float* gamma, float* Nw) {
  __shared__ float wsh[Tt];
  const int b = blockIdx.x, tid = threadIdx.x;
  for (int i = tid; i < Tt; i += 256) wsh[i] = w[b * Tt + i];
  __syncthreads();
  if (tid == 0) {
    float s = 0.f;
    for (int t = 0; t < Tt; ++t) s += wsh[t];
    Nw[b] = s;
  }
  const float* xb = x + (size_t)b * Tt * Dm;
  float a0 = 0.f, a1 = 0.f;
  for (int t = 0; t < Tt; ++t) {
    float wv = wsh[t];
    a0 += xb[t * Dm + tid] * wv;
    a1 += xb[t * Dm + tid + 256] * wv;
  }
  gamma[b * Dm + tid] = a0;
  gamma[b * Dm + tid + 256] = a1;
}

// cov_i = chol^T diag(var) chol : one 16x16 tile per wave, K=512 WMMA loop
__global__ void k_covi(const float* __restrict__ cholW,
                       const float* __restrict__ var, float* covi) {
  const int wid = (blockIdx.x * blockDim.x + threadIdx.x) >> 5;  // 0..1023
  const int ti = wid >> 5, tj = wid & 31;
  v8f acc = vzero();
  for (int kb = 0; kb < NB; ++kb) {
    const int K0 = kb * 16;
    acc = wmma16(
        [&](int m, int k) { return cholW[(K0 + k) * Dm + ti * 16 + m]; },
        [&](int k, int n) { return var[K0 + k] * cholW[(K0 + k) * Dm + tj * 16 + n]; },
        acc);
  }
  const int hi8 = ((threadIdx.x & 31) >> 4) << 3, ln = threadIdx.x & 15;
#pragma unroll
  for (int r = 0; r < 8; ++r)
    covi[(ti * 16 + r + hi8) * Dm + tj * 16 + ln] = acc[r];
}

// prec_i = inv(cov_i) = U0^T U0 via rchol + trtri on cov_i (single workgroup)
__global__ void k_prec(float* covw, float* preci) {
  __shared__ float pan[NB * 256];
  __shared__ float dg[16 * 17];
  __shared__ float dinv[16 * 17];
  __shared__ float stg[8 * 256];
  const int tid = threadIdx.x;
  rchol512(covw, pan, dg, dinv, tid);
  trtri512(covw, pan, stg, tid);
  const int wid = tid >> 5;
  const int hi8 = ((tid & 31) >> 4) << 3, ln = tid & 15;
  for (int t = wid; t < NB * (NB + 1) / 2; t += 8) {
    int j = 0;
    while ((j + 1) * (j + 2) / 2 <= t) ++j;
    int i = t - j * (j + 1) / 2;  // i <= j
    v8f acc = vzero();
    for (int k = 0; k <= i; ++k) {
      acc = wmma16(
          [&](int m, int kk) { return loadU(covw, pan, k * 16 + kk, i * 16 + m); },
          [&](int kk, int n) { return loadU(covw, pan, k * 16 + kk, j * 16 + n); },
          acc);
    }
#pragma unroll
    for (int r = 0; r < 8; ++r) {
      float v = acc[r];
      preci[(i * 16 + r + hi8) * Dm + j * 16 + ln] = v;
      preci[(j * 16 + ln) * Dm + i * 16 + r + hi8] = v;
    }
  }
}

// One workgroup per batch: build prec_b in the chol output slab, factor,
// invert, then emit mu / logvar / normalized upper-Cholesky in place.
__global__ void k_batch(const float* __restrict__ preci,
                        const float* __restrict__ Nw,
                        const float* __restrict__ gamma, float* mu,
                        float* logvar, float* cholO) {
  __shared__ float pan[NB * 256];  // rchol panel, then dall (inv diag blocks)
  __shared__ float dg[16 * 17];
  __shared__ float dinv[16 * 17];
  __shared__ float stg[8 * 256];
  __shared__ float gsh[Dm];
  __shared__ float tv[Dm];
  const int b = blockIdx.x, tid = threadIdx.x;
  float* Mb = cholO + (size_t)b * Dm * Dm;
  const float Nb = Nw[b];

  // prec_b = I + N_b * prec_i
  for (int idx = tid; idx < Dm * Dm; idx += 256) {
    int r = idx >> 9, c = idx & 511;
    float v = Nb * preci[idx];
    if (r == c) v += 1.0f;
    Mb[idx] = v;
  }
  __syncthreads();

  rchol512(Mb, pan, dg, dinv, tid);
  trtri512(Mb, pan, stg, tid);

  gsh[tid] = gamma[b * Dm + tid];
  gsh[tid + 256] = gamma[b * Dm + tid + 256];
  __syncthreads();

  // tv = U * gamma
  for (int rr = tid; rr < Dm; rr += 256) {
    float s = 0.f;
    for (int c = rr; c < Dm; ++c) s += loadU(Mb, pan, rr, c) * gsh[c];
    tv[rr] = s;
  }
  __syncthreads();

  // mu = U^T tv ; logvar = 2 log diag(U)
  for (int e = tid; e < Dm; e += 256) {
    float s = 0.f;
    for (int r = 0; r <= e; ++r) s += loadU(Mb, pan, r, e) * tv[r];
    mu[b * Dm + e] = s;
    float ud = pan[(e >> 4) * 256 + (e & 15) * 16 + (e & 15)];
    logvar[b * Dm + e] = 2.0f * logf(ud);
  }
  __syncthreads();

  // rewrite slab: unit-diagonal upper factor, zero below
  for (int r2 = tid; r2 < Dm; r2 += 256) Mb[r2 * Dm + r2] = 1.0f;
  for (int r = 0; r < Dm; ++r) {
    float invd = 1.0f / pan[(r >> 4) * 256 + (r & 15) * 16 + (r & 15)];
    for (int c = r + 1 + tid; c < Dm; c += 256) {
      float u = loadU(Mb, pan, r, c);
      Mb[r * Dm + c] = u * invd;
      Mb[c * Dm + r] = 0.f;
    }
  }
}

extern "C" void kernel_launch(void* const* d_in, const int* in_sizes, int n_in,
                              void* d_out, int out_size, void* d_ws,
                              size_t ws_size, hipStream_t stream) {
  (void)in_sizes; (void)n_in; (void)out_size; (void)ws_size;
  const float* x     = (const float*)d_in[0];
  const float* w     = (const float*)d_in[1];
  const float* Dw    = (const float*)d_in[2];
  const float* cholW = (const float*)d_in[3];

  float* out    = (float*)d_out;
  float* mu     = out;                       // [256,512]
  float* logvar = out + NBATCH * Dm;         // [256,512]
  float* cholO  = out + 2 * NBATCH * Dm;     // [256,512,512] (also workspace)

  float* ws    = (float*)d_ws;               // ~2.6 MB used
  float* gamma = ws;                         // 131072
  float* Nw    = gamma + NBATCH * Dm;        // 256
  float* var   = Nw + NBATCH;                // 512
  float* covi  = var + Dm;                   // 262144
  float* preci = covi + Dm * Dm;             // 262144

  k_var<<<1, 512, 0, stream>>>(Dw, var);
  k_stats<<<NBATCH, 256, 0, stream>>>(x, w, gamma, Nw);
  k_covi<<<256, 128, 0, stream>>>(cholW, var, covi);
  k_prec<<<1, 256, 0, stream>>>(covi, preci);
  k_batch<<<NBATCH, 256, 0, stream>>>(preci, Nw, gamma, mu, logvar, cholO);
}